// Relation_Classifier_24541443130054
// MI455X (gfx1250) — compile-verified
//
#include <hip/hip_runtime.h>

#define B_DIM 64
#define C_DIM 1000
#define D_DIM 2048
#define SCALE_CONF 10.0f
#define SCALE_COS  16.0f
#define KSPLIT 4

typedef __attribute__((ext_vector_type(2))) float v2f;
typedef __attribute__((ext_vector_type(8))) float v8f;

// Fragment bundle for one 16-wide K chunk (4 WMMA steps, 5 streams: A + 4 B).
// Only ever accessed with compile-time-constant indices -> stays in VGPRs.
struct Frag {
  v2f a[4], b0[4], b1[4], b2[4], b3[4];
};

__device__ __forceinline__ void load_chunk_nt(Frag& f, const float* ap,
                                              const float* w0, const float* w1,
                                              const float* w2, const float* w3, int k) {
#pragma unroll
  for (int u = 0; u < 4; ++u) {
    f.a[u]  = *(const v2f*)(ap + k + 4 * u);
    f.b0[u] = *(const v2f*)(w0 + k + 4 * u);
    f.b1[u] = *(const v2f*)(w1 + k + 4 * u);
    f.b2[u] = *(const v2f*)(w2 + k + 4 * u);
    f.b3[u] = *(const v2f*)(w3 + k + 4 * u);
  }
}

__device__ __forceinline__ void load_chunk_nn(Frag& f, const float* ap, const float* Bm,
                                              int N, int khi, int nc0, int nc1, int nc2,
                                              int nc3, int k) {
#pragma unroll
  for (int u = 0; u < 4; ++u) {
    f.a[u] = *(const v2f*)(ap + k + 4 * u);
    const float* r0 = Bm + (size_t)(k + 4 * u + khi) * N;
    const float* r1 = r0 + N;
    f.b0[u] = v2f{r0[nc0], r1[nc0]};
    f.b1[u] = v2f{r0[nc1], r1[nc1]};
    f.b2[u] = v2f{r0[nc2], r1[nc2]};
    f.b3[u] = v2f{r0[nc3], r1[nc3]};
  }
}

__device__ __forceinline__ void mma_chunk(const Frag& f, v8f& acc0, v8f& acc1,
                                          v8f& acc2, v8f& acc3) {
#pragma unroll
  for (int u = 0; u < 4; ++u) {
    acc0 = __builtin_amdgcn_wmma_f32_16x16x4_f32(false, f.a[u], false, f.b0[u], (short)0, acc0, false, false);
    acc1 = __builtin_amdgcn_wmma_f32_16x16x4_f32(false, f.a[u], false, f.b1[u], (short)0, acc1, false, false);
    acc2 = __builtin_amdgcn_wmma_f32_16x16x4_f32(false, f.a[u], false, f.b2[u], (short)0, acc2, false, false);
    acc3 = __builtin_amdgcn_wmma_f32_16x16x4_f32(false, f.a[u], false, f.b3[u], (short)0, acc3, false, false);
  }
}

__device__ __forceinline__ void store_tile(float* part, int M, int N, int mt, int nt,
                                           int lane, const v8f& acc0, const v8f& acc1,
                                           const v8f& acc2, const v8f& acc3) {
  const int lm = lane & 15;
  const int rowb = mt * 16 + ((lane >> 4) << 3);
#pragma unroll
  for (int r = 0; r < 8; ++r) {
    float* o = part + (size_t)(rowb + r) * N;
    int n = nt * 64 + lm;
    if (n < N) o[n] = acc0[r];
    n += 16; if (n < N) o[n] = acc1[r];
    n += 16; if (n < N) o[n] = acc2[r];
    n += 16; if (n < N) o[n] = acc3[r];
  }
}

// ---------------------------------------------------------------------------
// WMMA f32 GEMM (NT), K-split: Part[ks][M,N] = A[M,Kslice] @ W[:,Kslice]^T
// One wave: 16(M) x 64(N) strip; manually double-buffered 16-wide K chunks.
// Wave id goes through readfirstlane so all tile/loop control is SGPR-resident
// (scalar branches, EXEC provably all-ones for every WMMA).
// ---------------------------------------------------------------------------
__global__ __launch_bounds__(256) void gemm_nt_f32(const float* __restrict__ A,
                                                   const float* __restrict__ W,
                                                   float* __restrict__ Part,
                                                   int M, int N, int K, int S) {
  const int lane = threadIdx.x & 31;
  const int wave = __builtin_amdgcn_readfirstlane(
      blockIdx.x * (blockDim.x >> 5) + (threadIdx.x >> 5));
  const int ntiles = (N + 63) >> 6;
  const int mtiles = M >> 4;
  const int tilesPerSplit = mtiles * ntiles;
  const int ks = wave / tilesPerSplit;
  if (ks >= S) return;                              // scalar branch
  const int rem0 = wave - ks * tilesPerSplit;
  const int mt = rem0 / ntiles;
  const int nt = rem0 - mt * ntiles;

  const int kq = (((K + S - 1) / S) + 3) & ~3;      // 4-aligned split quantum
  const int kbeg = ks * kq;
  const int kend = (kbeg + kq < K) ? (kbeg + kq) : K;
  if (kbeg >= kend) return;                         // scalar branch

  const int lm  = lane & 15;
  const int khi = (lane >> 4) << 1;                 // 0 or 2
  const float* ap = A + (size_t)(mt * 16 + lm) * K + khi;
  const int n0 = nt * 64 + lm;
  const float* w0 = W + (size_t)(n0      < N ? n0      : N - 1) * K + khi;
  const float* w1 = W + (size_t)(n0 + 16 < N ? n0 + 16 : N - 1) * K + khi;
  const float* w2 = W + (size_t)(n0 + 32 < N ? n0 + 32 : N - 1) * K + khi;
  const float* w3 = W + (size_t)(n0 + 48 < N ? n0 + 48 : N - 1) * K + khi;

  v8f acc0 = {}, acc1 = {}, acc2 = {}, acc3 = {};
  Frag fA, fB;
  int k = kbeg;
  const int nchunks = (kend - kbeg) >> 4;
  int c = 0;
  if (nchunks > 0) {
    load_chunk_nt(fA, ap, w0, w1, w2, w3, k);
    while (c + 1 < nchunks) {                       // steady state: 2 chunks/iter
      load_chunk_nt(fB, ap, w0, w1, w2, w3, k + 16);
      mma_chunk(fA, acc0, acc1, acc2, acc3);
      k += 16; ++c;
      if (c + 1 < nchunks) load_chunk_nt(fA, ap, w0, w1, w2, w3, k + 16);
      mma_chunk(fB, acc0, acc1, acc2, acc3);
      k += 16; ++c;
    }
    if (c < nchunks) {                              // last resident chunk (in fA)
      mma_chunk(fA, acc0, acc1, acc2, acc3);
      k += 16; ++c;
    }
  }
  for (; k < kend; k += 4) {                        // <=3 leftover 4-steps
    v2f a  = *(const v2f*)(ap + k);
    v2f b0 = *(const v2f*)(w0 + k);
    v2f b1 = *(const v2f*)(w1 + k);
    v2f b2 = *(const v2f*)(w2 + k);
    v2f b3 = *(const v2f*)(w3 + k);
    acc0 = __builtin_amdgcn_wmma_f32_16x16x4_f32(false, a, false, b0, (short)0, acc0, false, false);
    acc1 = __builtin_amdgcn_wmma_f32_16x16x4_f32(false, a, false, b1, (short)0, acc1, false, false);
    acc2 = __builtin_amdgcn_wmma_f32_16x16x4_f32(false, a, false, b2, (short)0, acc2, false, false);
    acc3 = __builtin_amdgcn_wmma_f32_16x16x4_f32(false, a, false, b3, (short)0, acc3, false, false);
  }
  store_tile(Part + (size_t)ks * M * N, M, N, mt, nt, lane, acc0, acc1, acc2, acc3);
}

// ---------------------------------------------------------------------------
// WMMA f32 GEMM (NN), K-split: Part[ks][M,N] = A[M,Kslice] @ Bm[Kslice,N]
// ---------------------------------------------------------------------------
__global__ __launch_bounds__(256) void gemm_nn_f32(const float* __restrict__ A,
                                                   const float* __restrict__ Bm,
                                                   float* __restrict__ Part,
                                                   int M, int N, int K, int S) {
  const int lane = threadIdx.x & 31;
  const int wave = __builtin_amdgcn_readfirstlane(
      blockIdx.x * (blockDim.x >> 5) + (threadIdx.x >> 5));
  const int ntiles = (N + 63) >> 6;
  const int mtiles = M >> 4;
  const int tilesPerSplit = mtiles * ntiles;
  const int ks = wave / tilesPerSplit;
  if (ks >= S) return;
  const int rem0 = wave - ks * tilesPerSplit;
  const int mt = rem0 / ntiles;
  const int nt = rem0 - mt * ntiles;

  const int kq = (((K + S - 1) / S) + 3) & ~3;
  const int kbeg = ks * kq;
  const int kend = (kbeg + kq < K) ? (kbeg + kq) : K;
  if (kbeg >= kend) return;

  const int lm  = lane & 15;
  const int khi = (lane >> 4) << 1;
  const float* ap = A + (size_t)(mt * 16 + lm) * K + khi;
  const int n0 = nt * 64 + lm;
  const int nc0 = (n0      < N) ? n0      : N - 1;
  const int nc1 = (n0 + 16 < N) ? n0 + 16 : N - 1;
  const int nc2 = (n0 + 32 < N) ? n0 + 32 : N - 1;
  const int nc3 = (n0 + 48 < N) ? n0 + 48 : N - 1;

  v8f acc0 = {}, acc1 = {}, acc2 = {}, acc3 = {};
  Frag fA, fB;
  int k = kbeg;
  const int nchunks = (kend - kbeg) >> 4;
  int c = 0;
  if (nchunks > 0) {
    load_chunk_nn(fA, ap, Bm, N, khi, nc0, nc1, nc2, nc3, k);
    while (c + 1 < nchunks) {
      load_chunk_nn(fB, ap, Bm, N, khi, nc0, nc1, nc2, nc3, k + 16);
      mma_chunk(fA, acc0, acc1, acc2, acc3);
      k += 16; ++c;
      if (c + 1 < nchunks) load_chunk_nn(fA, ap, Bm, N, khi, nc0, nc1, nc2, nc3, k + 16);
      mma_chunk(fB, acc0, acc1, acc2, acc3);
      k += 16; ++c;
    }
    if (c < nchunks) {
      mma_chunk(fA, acc0, acc1, acc2, acc3);
      k += 16; ++c;
    }
  }
  for (; k < kend; k += 4) {
    v2f a = *(const v2f*)(ap + k);
    const float* r0 = Bm + (size_t)(k + khi) * N;
    const float* r1 = r0 + N;
    v2f b0 = v2f{r0[nc0], r1[nc0]};
    v2f b1 = v2f{r0[nc1], r1[nc1]};
    v2f b2 = v2f{r0[nc2], r1[nc2]};
    v2f b3 = v2f{r0[nc3], r1[nc3]};
    acc0 = __builtin_amdgcn_wmma_f32_16x16x4_f32(false, a, false, b0, (short)0, acc0, false, false);
    acc1 = __builtin_amdgcn_wmma_f32_16x16x4_f32(false, a, false, b1, (short)0, acc1, false, false);
    acc2 = __builtin_amdgcn_wmma_f32_16x16x4_f32(false, a, false, b2, (short)0, acc2, false, false);
    acc3 = __builtin_amdgcn_wmma_f32_16x16x4_f32(false, a, false, b3, (short)0, acc3, false, false);
  }
  store_tile(Part + (size_t)ks * M * N, M, N, mt, nt, lane, acc0, acc1, acc2, acc3);
}

// ---------------------------------------------------------------------------
// Sum the K-split partial planes: out[i] = sum_s part[s][i]
// ---------------------------------------------------------------------------
__global__ __launch_bounds__(256) void reduce_ksplit(const float* __restrict__ part,
                                                     float* __restrict__ out,
                                                     int MN, int S) {
  const int i = blockIdx.x * 256 + threadIdx.x;
  if (i >= MN) return;
  float s = 0.f;
  for (int j = 0; j < S; ++j) s += part[(size_t)j * MN + i];
  out[i] = s;
}

// ---------------------------------------------------------------------------
// Row sum-of-squares: out[r] = sum_d in[r,d]^2
// ---------------------------------------------------------------------------
__global__ __launch_bounds__(256) void row_sumsq(const float* __restrict__ in,
                                                 float* __restrict__ out, int D) {
  const int r = blockIdx.x;
  const float* p = in + (size_t)r * D;
  float s = 0.f;
  for (int i = threadIdx.x; i < D; i += 256) { float v = p[i]; s += v * v; }
  __shared__ float lds[256];
  lds[threadIdx.x] = s; __syncthreads();
  for (int off = 128; off > 0; off >>= 1) {
    if (threadIdx.x < off) lds[threadIdx.x] += lds[threadIdx.x + off];
    __syncthreads();
  }
  if (threadIdx.x == 0) out[r] = lds[0];
}

// ---------------------------------------------------------------------------
// Per-row: confidence = 10/sqrt(min_c max(x2 - 2*xc + c2, 0)),
//          prob = softmax(logits1 + fc1_b)  (in place)
// ---------------------------------------------------------------------------
__global__ __launch_bounds__(256) void row_ops(const float* __restrict__ xc,
                                               float* __restrict__ l1prob,
                                               const float* __restrict__ fc1_b,
                                               const float* __restrict__ x2,
                                               const float* __restrict__ c2,
                                               float* __restrict__ conf, int C) {
  const int b = blockIdx.x, t = threadIdx.x;
  const float* xcr = xc + (size_t)b * C;
  float* lr = l1prob + (size_t)b * C;
  const float x2b = x2[b];
  float mind = 3.4e38f, maxl = -3.4e38f;
  for (int c = t; c < C; c += 256) {
    float d2 = fmaxf(x2b - 2.f * xcr[c] + c2[c], 0.f);
    mind = fminf(mind, d2);
    maxl = fmaxf(maxl, lr[c] + fc1_b[c]);
  }
  __shared__ float lmin[256], lmax[256], lsum[256];
  lmin[t] = mind; lmax[t] = maxl; __syncthreads();
  for (int off = 128; off > 0; off >>= 1) {
    if (t < off) {
      lmin[t] = fminf(lmin[t], lmin[t + off]);
      lmax[t] = fmaxf(lmax[t], lmax[t + off]);
    }
    __syncthreads();
  }
  mind = lmin[0]; maxl = lmax[0];
  float ssum = 0.f;
  for (int c = t; c < C; c += 256) {
    float e = __expf(lr[c] + fc1_b[c] - maxl);
    lr[c] = e; ssum += e;
  }
  lsum[t] = ssum; __syncthreads();
  for (int off = 128; off > 0; off >>= 1) {
    if (t < off) lsum[t] += lsum[t + off];
    __syncthreads();
  }
  const float inv = 1.f / lsum[0];
  for (int c = t; c < C; c += 256) lr[c] *= inv;
  if (t == 0) conf[b] = SCALE_CONF / sqrtf(mind);
}

// ---------------------------------------------------------------------------
// Elementwise fuse: selector = tanh(chs + b); fast = selector * centers_cur;
// x_new = conf * (x + fast); emit slow/fast outputs.
// ---------------------------------------------------------------------------
__global__ __launch_bounds__(256) void fuse_eltwise(const float* __restrict__ x,
                                                    const float* __restrict__ chs,
                                                    const float* __restrict__ fcb,
                                                    const float* __restrict__ ccur,
                                                    const float* __restrict__ conf,
                                                    float* __restrict__ slow,
                                                    float* __restrict__ fast,
                                                    float* __restrict__ xnew,
                                                    int D, int total) {
  const int i = blockIdx.x * 256 + threadIdx.x;
  if (i >= total) return;
  const int b = i / D, d = i - b * D;
  const float xv = x[i];
  const float sel = tanhf(chs[i] + fcb[d]);
  const float fv = sel * ccur[i];
  slow[i] = xv;
  fast[i] = fv;
  xnew[i] = conf[b] * (xv + fv);
}

// ---------------------------------------------------------------------------
// CosNorm epilogue: logits[b,c] = 16/(1+||x_new_b||) * g[b,c] / ||cos_w_c||
// ---------------------------------------------------------------------------
__global__ __launch_bounds__(256) void final_scale(const float* __restrict__ g,
                                                   const float* __restrict__ xnsq,
                                                   const float* __restrict__ wn2,
                                                   float* __restrict__ logits,
                                                   int C, int total) {
  const int i = blockIdx.x * 256 + threadIdx.x;
  if (i >= total) return;
  const int b = i / C, c = i - b * C;
  const float s = SCALE_COS / (1.0f + sqrtf(xnsq[b]));
  logits[i] = s * g[i] / sqrtf(wn2[c]);
}

// ---------------------------------------------------------------------------
extern "C" void kernel_launch(void* const* d_in, const int* in_sizes, int n_in,
                              void* d_out, int out_size, void* d_ws, size_t ws_size,
                              hipStream_t stream) {
  (void)in_sizes; (void)n_in; (void)out_size; (void)ws_size;
  const float* x       = (const float*)d_in[0];
  // d_in[1] = labels (unused), d_in[3] = class_count (unused)
  const float* centers = (const float*)d_in[2];
  const float* fccw    = (const float*)d_in[4];
  const float* fccb    = (const float*)d_in[5];
  const float* fc1w    = (const float*)d_in[6];
  const float* fc1b    = (const float*)d_in[7];
  const float* cosw    = (const float*)d_in[8];

  float* out_logits = (float*)d_out;                       // [B,C]
  float* out_slow   = out_logits + B_DIM * C_DIM;          // [B,D]
  float* out_fast   = out_slow + B_DIM * D_DIM;            // [B,D]

  float* ws   = (float*)d_ws;
  float* xc   = ws;                  // 64000
  float* prob = ws + 64000;          // 64000 (logits1 -> prob, in place)
  float* c2   = ws + 128000;         // 1000
  float* x2   = ws + 129000;         // 64
  float* wn2  = ws + 129064;         // 1000
  float* conf = ws + 130064;         // 64
  float* ccur = ws + 130128;         // 131072
  float* chs  = ws + 261200;         // 131072
  float* xnew = ws + 392272;         // 131072
  float* g    = ws + 523344;         // 64000
  float* xnsq = ws + 587344;         // 64
  float* part = ws + 587408;         // KSPLIT * 64 * 2048 = 524288 (reused per GEMM)

  dim3 blk(256);
  const int S = KSPLIT;
  auto gemm_blocks = [S](int M, int N) {
    int waves = (M >> 4) * ((N + 63) >> 6) * S;
    return (waves + 7) / 8;
  };
  const int BC = B_DIM * C_DIM;   // 64000
  const int BD = B_DIM * D_DIM;   // 131072
  const int rbBC = (BC + 255) / 256;
  const int rbBD = (BD + 255) / 256;

  // row norms (independent)
  row_sumsq<<<B_DIM, blk, 0, stream>>>(x, x2, D_DIM);
  row_sumsq<<<C_DIM, blk, 0, stream>>>(centers, c2, D_DIM);
  row_sumsq<<<C_DIM, blk, 0, stream>>>(cosw, wn2, D_DIM);

  // stage-1 GEMMs (K-split partials + reduce)
  gemm_nt_f32<<<gemm_blocks(B_DIM, C_DIM), blk, 0, stream>>>(x, centers, part, B_DIM, C_DIM, D_DIM, S);
  reduce_ksplit<<<rbBC, blk, 0, stream>>>(part, xc, BC, S);
  gemm_nt_f32<<<gemm_blocks(B_DIM, C_DIM), blk, 0, stream>>>(x, fc1w, part, B_DIM, C_DIM, D_DIM, S);
  reduce_ksplit<<<rbBC, blk, 0, stream>>>(part, prob, BC, S);

  // confidence + softmax
  row_ops<<<B_DIM, blk, 0, stream>>>(xc, prob, fc1b, x2, c2, conf, C_DIM);

  // memory hallucination + channel selector GEMMs
  gemm_nn_f32<<<gemm_blocks(B_DIM, D_DIM), blk, 0, stream>>>(prob, centers, part, B_DIM, D_DIM, C_DIM, S);
  reduce_ksplit<<<rbBD, blk, 0, stream>>>(part, ccur, BD, S);
  gemm_nt_f32<<<gemm_blocks(B_DIM, D_DIM), blk, 0, stream>>>(x, fccw, part, B_DIM, D_DIM, D_DIM, S);
  reduce_ksplit<<<rbBD, blk, 0, stream>>>(part, chs, BD, S);

  // fuse elementwise, produce slow/fast outputs + x_new
  fuse_eltwise<<<rbBD, blk, 0, stream>>>(x, chs, fccb, ccur, conf,
                                         out_slow, out_fast, xnew, D_DIM, BD);

  // CosNorm classifier
  row_sumsq<<<B_DIM, blk, 0, stream>>>(xnew, xnsq, D_DIM);
  gemm_nt_f32<<<gemm_blocks(B_DIM, C_DIM), blk, 0, stream>>>(xnew, cosw, part, B_DIM, C_DIM, D_DIM, S);
  reduce_ksplit<<<rbBC, blk, 0, stream>>>(part, g, BC, S);
  final_scale<<<rbBC, blk, 0, stream>>>(g, xnsq, wn2, out_logits, C_DIM, BC);
}